// GenAttentionMask_26414048871073
// MI455X (gfx1250) — compile-verified
//
#include <hip/hip_runtime.h>

// GenAttentionMask for MI455X (gfx1250): pure data movement, driven entirely by
// the CDNA5 Tensor Data Mover.
//   HBM traffic ~= 381MB stores + 24MB unique reads  ->  ~17us floor @ 23.3TB/s.
//   TDM does the strided 2D row-gather (handles odd s_i alignment natively) and
//   16 contiguous DMA stores per tile; the wave only builds descriptors.

typedef unsigned int       u32;
typedef unsigned long long u64;
typedef _Float16           f16;

typedef __attribute__((ext_vector_type(4))) u32 v4u;
typedef __attribute__((ext_vector_type(8))) int v8i;
typedef __attribute__((ext_vector_type(4))) int v4i;

#define SDIM      2048          // mask row stride (reference S)
#define MAX_WIN   256           // max row-windows per sample (covers s=2047, 8 rows/window)
#define LDS_BYTES 32768         // dynamic LDS per block
#define LDS_ELEMS (LDS_BYTES / 2)

// ---- D# group 0 (128b): count=1 | lds_addr | global_addr[56:0] | type=2 ----
__device__ __forceinline__ v4u tdm_g0(u32 lds_off, u64 gaddr) {
  v4u g;
  g.x = 1u;                                                   // count=1, user mode, no gather
  g.y = lds_off;                                              // lds_addr (bytes)
  g.z = (u32)gaddr;                                           // global_addr[31:0]
  g.w = (u32)((gaddr >> 32) & 0x01FFFFFFu) | (2u << 30);      // global_addr[56:32] | type=2
  return g;
}

// ---- D# group 1 (256b): data_size=2B; dims/tiles/stride per ISA 8.4 ----
__device__ __forceinline__ v8i tdm_g1(u32 tensor_d0, u32 tensor_d1,
                                      u32 tile_d0, u32 tile_d1, u64 stride0) {
  v8i g;
  g[0] = (int)(1u << 16);                                     // wg_mask=0, data_size=1 (2 bytes)
  g[1] = (int)((tensor_d0 & 0xFFFFu) << 16);                  // abar_addr=0 | tensor_dim0[15:0]
  g[2] = (int)((tensor_d0 >> 16) | ((tensor_d1 & 0xFFFFu) << 16));
  g[3] = (int)((tensor_d1 >> 16) | ((tile_d0 & 0xFFFFu) << 16));
  g[4] = (int)(tile_d1 & 0xFFFFu);                            // tile_dim1 | tile_dim2=0
  g[5] = (int)((u32)stride0);                                 // tensor_dim0_stride[31:0]
  g[6] = (int)((u32)((stride0 >> 32) & 0xFFFFu));             // stride0[47:32] | stride1[15:0]=0
  g[7] = 0;                                                   // stride1[47:16]=0
  return g;
}

__global__ __launch_bounds__(32)
void genmask_tdm_kernel(const f16* __restrict__ mask,
                        const int* __restrict__ seq_len,
                        const int* __restrict__ num_heads_p,
                        f16* __restrict__ out, int Bn)
{
  // One wave32 per block; all values below are wave-uniform -> SGPR descriptors.
  const int bid = (int)blockIdx.x;
  const int i   = bid / MAX_WIN;        // sample
  const int w   = bid % MAX_WIN;        // row-window within sample
  if (i >= Bn) return;

  const int s = seq_len[i];
  if (s <= 0) return;
  const int H = num_heads_p[0];

  int rpw = LDS_ELEMS / s;              // full rows per 32KB window
  if (rpw > s) rpw = s;
  const int r0 = w * rpw;
  if (r0 >= s) return;                  // overlaunched window: idle
  int rows = s - r0;
  if (rows > rpw) rows = rpw;

  // Packed output base for this sample: H * sum_{j<i} s_j^2
  u64 base = 0;
  for (int j = 0; j < i; ++j) { u64 sj = (u64)seq_len[j]; base += sj * sj; }
  base *= (u64)H;
  const u64 piece = (u64)s * (u64)s;

  // No static LDS in this kernel -> dynamic LDS segment begins at offset 0.
  const u32 lds_off = 0;
  const v4i z4 = {0, 0, 0, 0};                   // D# groups 2/3 unused (<=2D tensors)
  const v8i z8 = {0, 0, 0, 0, 0, 0, 0, 0};       // trailing zero operand (6-arg builtin form)

  // ---- TDM gather: rows x s tile of mask[i], row stride SDIM, compacted to LDS ----
  {
    const u64 g_src = (u64)(mask + (u64)i * SDIM * SDIM + (u64)r0 * SDIM);
    v4u g0 = tdm_g0(lds_off, g_src);
    v8i g1 = tdm_g1(/*tensor_d0=*/(u32)s, /*tensor_d1=*/(u32)(s - r0),
                    /*tile_d0=*/(u32)s,  /*tile_d1=*/(u32)rows,
                    /*stride0=*/(u64)SDIM);
    __builtin_amdgcn_tensor_load_to_lds(g0, g1, z4, z4, z8, 0);
  }
  __builtin_amdgcn_s_wait_tensorcnt(0); // LDS tile ready

  // ---- TDM scatter: H contiguous 1D stores of the packed tile ----
  const u32 L = (u32)rows * (u32)s;     // <= 16384, fits 16-bit tile_dim0
  u64 doff = base + (u64)r0 * (u64)s;   // head-0 destination (elements)
  for (int h = 0; h < H; ++h) {
    const u64 g_dst = (u64)(out + doff);
    v4u g0 = tdm_g0(lds_off, g_dst);
    v8i g1 = tdm_g1(/*tensor_d0=*/L, /*tensor_d1=*/1u,
                    /*tile_d0=*/L,  /*tile_d1=*/1u,
                    /*stride0=*/(u64)L);
    __builtin_amdgcn_tensor_store_from_lds(g0, g1, z4, z4, z8, 0);
    doff += piece;
  }
  // S_ENDPGM performs an implicit wait-idle, covering outstanding tensor stores.
}

extern "C" void kernel_launch(void* const* d_in, const int* in_sizes, int n_in,
                              void* d_out, int out_size, void* d_ws, size_t ws_size,
                              hipStream_t stream) {
  (void)n_in; (void)out_size; (void)d_ws; (void)ws_size;
  const f16* mask     = (const f16*)d_in[0];   // fp16 [8, 2048, 2048]
  const int* seq_len  = (const int*)d_in[1];   // int32 [8]
  const int* num_head = (const int*)d_in[2];   // int32 scalar (=16)
  f16*       out      = (f16*)d_out;           // fp16 packed output

  const int Bn = in_sizes[1];                  // batch (=8)
  const int blocks = Bn * MAX_WIN;             // 2048 one-wave workgroups

  genmask_tdm_kernel<<<dim3(blocks), dim3(32), LDS_BYTES, stream>>>(
      mask, seq_len, num_head, out, Bn);
}